// PINN_KDV_17832704213053
// MI455X (gfx1250) — compile-verified
//
#include <hip/hip_runtime.h>
#include <math.h>

// ---- problem constants (match reference) ----
#define QQ   50
#define HH   50
#define NHID 4
#define NLAY 5          // 4 hidden + output (all with tanh)
#define DTc  0.6f
#define LBc  (-1.0f)
#define UBc  (1.0f)

// ---- tiling ----
#define BW    8          // waves per block
#define BLOCK (BW*32)
#define KP    64         // padded neuron dim (50 -> 64)
#define WP2ST 160        // dword stride per k2 row-pair (bank-conflict-free b64 reads)
#define AST   68         // activation scratch row stride (bank-conflict-free A reads)

typedef __attribute__((ext_vector_type(2))) float v2f;
typedef __attribute__((ext_vector_type(8))) float v8f;

__device__ __forceinline__ v8f wmma4(v2f a, v2f b, v8f c) {
  // D(16x16,f32) = A(16x4,f32) * B(4x16,f32) + C
  return __builtin_amdgcn_wmma_f32_16x16x4_f32(false, a, false, b, (short)0, c,
                                               false, false);
}

__device__ __forceinline__ float htanh(float x) {
#if __has_builtin(__builtin_amdgcn_tanhf)
  return __builtin_amdgcn_tanhf(x);     // gfx1250 V_TANH_F32 (TRANS, co-executes)
#else
  return tanhf(x);
#endif
}

// raw-derivative jet through tanh (Faa di Bruno up to 3rd order)
__device__ __forceinline__ void tanh_jet(float z0, float z1, float z2, float z3,
                                         float& a0, float& a1, float& a2, float& a3) {
  float t  = htanh(z0);
  float t1 = 1.0f - t * t;               // f'
  float t2 = -2.0f * t * t1;             // f''
  float t3 = -2.0f * (t1 * t1 + t * t2); // f'''
  a0 = t;
  a1 = t1 * z1;
  a2 = t2 * z1 * z1 + t1 * z2;
  a3 = t3 * z1 * z1 * z1 + 3.0f * t2 * z1 * z2 + t1 * z3;
}

__global__ void zero_out_k(float* o) { *o = 0.0f; }

__global__ __launch_bounds__(BLOCK, 1)
void pinn_kdv(const float* __restrict__ X1, const float* __restrict__ X2,
              const float* __restrict__ Win, const float* __restrict__ bin,
              const float* __restrict__ Whid, const float* __restrict__ bhid,
              const float* __restrict__ Wout, const float* __restrict__ bout,
              const float* __restrict__ lam1p, const float* __restrict__ lam2p,
              const float* __restrict__ alpha, const float* __restrict__ beta,
              float* __restrict__ out, int Nn)
{
  __shared__ float sWin[KP], sBin[KP];
  __shared__ float sWp[NLAY][32 * WP2ST]; // row-pair interleaved: [k/2][j*2 + (k&1)]
  __shared__ float sB[NLAY][KP];
  __shared__ float sMp[32 * WP2ST];       // IRK matrix^T, same pairing: [j/2][q*2+(j&1)]
  __shared__ float sAct[BW][4][16][AST];  // per-wave jet scratch [jet][sample][neuron]

  const int tid    = threadIdx.x;
  const int branch = blockIdx.y;          // 0: x1/alpha, 1: x2/(beta-alpha)

  // ---- cooperative preload of padded weights into LDS ----
  for (int i = tid; i < KP; i += BLOCK) {
    sWin[i] = (i < HH) ? Win[i] : 0.0f;
    sBin[i] = (i < HH) ? bin[i] : 0.0f;
  }
  for (int i = tid; i < NLAY * KP; i += BLOCK) {
    int L = i >> 6, j = i & 63;
    const float* bb = (L < NHID) ? (bhid + L * HH) : bout;
    int cols = (L < NHID) ? HH : QQ;
    sB[L][j] = (j < cols) ? bb[j] : 0.0f;
  }
  for (int i = tid; i < NLAY * KP * KP; i += BLOCK) {
    int L = i >> 12, k = (i >> 6) & 63, j = i & 63;
    const float* Wm = (L < NHID) ? (Whid + L * HH * HH) : Wout;
    int cols = (L < NHID) ? HH : QQ;
    float v = (k < HH && j < cols) ? Wm[k * cols + j] : 0.0f;
    sWp[L][(k >> 1) * WP2ST + j * 2 + (k & 1)] = v;
  }
  for (int i = tid; i < KP * KP; i += BLOCK) {
    int j = i >> 6, q = i & 63;
    float v = 0.0f;
    if (j < QQ && q < QQ) {
      float av = alpha[q * QQ + j];       // alpha[q][j]
      v = branch ? (beta[j] - av) : av;   // (beta - alpha).T or alpha.T
    }
    sMp[(j >> 1) * WP2ST + q * 2 + (j & 1)] = v;
  }
  __syncthreads();

  const int lane = tid & 31;
  const int wv   = tid >> 5;
  const int n16  = lane & 15;
  const int hi   = lane >> 4;     // half-wave
  const int khi  = hi << 1;       // K offset (+2) for A/B upper half
  const int mhi  = hi << 3;       // M offset (+8) for C/D upper half

  const float lam1 = lam1p[0];
  const float lam2 = expf(lam2p[0]);
  const float* X   = branch ? X2 : X1;
  const float sx   = 2.0f / (UBc - LBc);

  float s2 = 0.0f;
  const int ntiles = (Nn + 15) >> 4;

  for (int tile = blockIdx.x * BW + wv; tile < ntiles; tile += gridDim.x * BW) {
    // ---- input layer (1 -> 64 padded), jets computed scalarly into LDS ----
    {
      int m = lane >> 1;
      int s = tile * 16 + m;
      float x  = (s < Nn) ? X[s] : 0.0f;
      float h0 = (x - LBc) * sx - 1.0f;  // == x for [-1,1]
      float h1 = sx;                     // dh/dx
      int jb = (lane & 1) * 32;
      #pragma unroll
      for (int e = 0; e < 32; ++e) {
        int j = jb + e;
        float wvl = sWin[j];
        float z0 = h0 * wvl + sBin[j];
        float z1 = h1 * wvl;
        float a0, a1, a2, a3;
        tanh_jet(z0, z1, 0.0f, 0.0f, a0, a1, a2, a3);
        sAct[wv][0][m][j] = a0; sAct[wv][1][m][j] = a1;
        sAct[wv][2][m][j] = a2; sAct[wv][3][m][j] = a3;
      }
    }

    // ---- 4 hidden layers + output layer: 4 jets x 4 N-tiles, K-chunk outer ----
    for (int L = 0; L < NLAY; ++L) {
      v8f acc[4][4];
      #pragma unroll
      for (int t = 0; t < 4; ++t) {
        float bvv = sB[L][t * 16 + n16];
        v8f c0, zr;
        #pragma unroll
        for (int r = 0; r < 8; ++r) { c0[r] = bvv; zr[r] = 0.0f; }
        acc[0][t] = c0; acc[1][t] = zr; acc[2][t] = zr; acc[3][t] = zr;
      }
      #pragma unroll
      for (int c4 = 0; c4 < 16; ++c4) {
        int kb = c4 * 4 + khi;            // even K base for this lane half
        int k2 = c4 * 2 + hi;             // row-pair index
        const v2f* wrow = (const v2f*)&sWp[L][k2 * WP2ST];
        v2f bf[4], af[4];
        #pragma unroll
        for (int t = 0; t < 4; ++t) bf[t] = wrow[t * 16 + n16];          // 1x b64 each
        #pragma unroll
        for (int k = 0; k < 4; ++k)
          af[k] = *(const v2f*)&sAct[wv][k][n16][kb];                    // 1x b64 each
        #pragma unroll
        for (int k = 0; k < 4; ++k) {
          #pragma unroll
          for (int t = 0; t < 4; ++t) {
            acc[k][t] = wmma4(af[k], bf[t], acc[k][t]);
          }
        }
      }
      if (L < NLAY - 1) {           // tanh jets, write back to scratch
        #pragma unroll
        for (int t = 0; t < 4; ++t) {
          #pragma unroll
          for (int r = 0; r < 8; ++r) {
            float a0, a1, a2, a3;
            tanh_jet(acc[0][t][r], acc[1][t][r], acc[2][t][r], acc[3][t][r],
                     a0, a1, a2, a3);
            int m = r + mhi, j = t * 16 + n16;
            sAct[wv][0][m][j] = a0; sAct[wv][1][m][j] = a1;
            sAct[wv][2][m][j] = a2; sAct[wv][3][m][j] = a3;
          }
        }
      } else {                      // output tanh, then N = -l1*U*U_x - l2*U_xxx
        #pragma unroll
        for (int t = 0; t < 4; ++t) {
          #pragma unroll
          for (int r = 0; r < 8; ++r) {
            float a0, a1, a2, a3;
            tanh_jet(acc[0][t][r], acc[1][t][r], acc[2][t][r], acc[3][t][r],
                     a0, a1, a2, a3);
            int m = r + mhi, j = t * 16 + n16;
            float ns = -lam1 * a0 * a1 - lam2 * a3;
            float valid = (tile * 16 + m < Nn) ? 1.0f : 0.0f;
            sAct[wv][0][m][j] = ns * valid;
          }
        }
      }
    }

    // ---- IRK contraction G = N @ M^T, accumulate ||G||^2 ----
    {
      v8f g[4];
      #pragma unroll
      for (int t = 0; t < 4; ++t) {
        #pragma unroll
        for (int r = 0; r < 8; ++r) g[t][r] = 0.0f;
      }
      #pragma unroll
      for (int c4 = 0; c4 < 16; ++c4) {
        int kb = c4 * 4 + khi;
        int k2 = c4 * 2 + hi;
        const v2f* mrow = (const v2f*)&sMp[k2 * WP2ST];
        v2f af = *(const v2f*)&sAct[wv][0][n16][kb];
        #pragma unroll
        for (int t = 0; t < 4; ++t) {
          v2f bf = mrow[t * 16 + n16];
          g[t] = wmma4(af, bf, g[t]);
        }
      }
      #pragma unroll
      for (int t = 0; t < 4; ++t) {
        #pragma unroll
        for (int r = 0; r < 8; ++r) s2 += g[t][r] * g[t][r];
      }
    }
  }

  // ---- wave reduction + global accumulate (loss scaled by DT^2) ----
  #pragma unroll
  for (int off = 16; off > 0; off >>= 1) s2 += __shfl_xor(s2, off, 32);
  if (lane == 0) atomicAdd(out, s2 * (DTc * DTc));
}

extern "C" void kernel_launch(void* const* d_in, const int* in_sizes, int n_in,
                              void* d_out, int out_size, void* d_ws, size_t ws_size,
                              hipStream_t stream) {
  const float* x1    = (const float*)d_in[0];
  const float* x2    = (const float*)d_in[1];
  const float* Win   = (const float*)d_in[2];
  const float* bin   = (const float*)d_in[3];
  const float* Whid  = (const float*)d_in[4];
  const float* bhid  = (const float*)d_in[5];
  const float* Wout  = (const float*)d_in[6];
  const float* bout  = (const float*)d_in[7];
  const float* lam1  = (const float*)d_in[8];
  const float* lam2  = (const float*)d_in[9];
  const float* alpha = (const float*)d_in[10];
  const float* beta  = (const float*)d_in[11];
  float* out = (float*)d_out;
  int Nn = in_sizes[0];

  zero_out_k<<<1, 1, 0, stream>>>(out);
  dim3 grid(256, 2, 1);   // x: sample-tile blocks, y: branch (x1 / x2)
  pinn_kdv<<<grid, BLOCK, 0, stream>>>(x1, x2, Win, bin, Whid, bhid, Wout, bout,
                                       lam1, lam2, alpha, beta, out, Nn);
}